// LabelNet_79164837200440
// MI455X (gfx1250) — compile-verified
//
#include <hip/hip_runtime.h>
#include <cstdint>
#include <cstddef>

// Problem sizes (match reference).
#define N1 1000000
#define CC 48
#define N2 131072
#define N3 16384
#define N4 2048
#define N5 256

// Workspace layout (bytes).
#define F2_BYTES (N2 * CC * 4)              // 25165824
#define F3_BYTES (N3 * CC * 4)              //  3145728
#define F4_BYTES (N4 * CC * 4)              //   393216
#define F5_BYTES (N5 * CC * 4)              //    49152
#define WS_TOTAL (F2_BYTES + F3_BYTES + F4_BYTES + F5_BYTES)

typedef __attribute__((ext_vector_type(2))) float v2f;
typedef __attribute__((ext_vector_type(8))) float v8f;

// ---------------------------------------------------------------------------
// Level-1 scatter: slabel[1M x 48] += into f2[131072 x 48] via parent2.
// Streams slabel through LDS using CDNA5 async global->LDS b128 loads
// (ASYNCcnt path), then issues 4 hardware f32 atomics per 16B chunk.
// Grid covers exactly N1*48/4 = 12,000,000 quads -> EXEC all ones.
// ---------------------------------------------------------------------------
__global__ void __launch_bounds__(256) scatter_l1(const float* __restrict__ slabel,
                                                  const int* __restrict__ parent,
                                                  float* __restrict__ fdst) {
    __shared__ float4 stage[256];
    const int gid = blockIdx.x * 256 + threadIdx.x;   // quad index, < 12M
    const float* gsrc = slabel + (size_t)gid * 4;

    // Low 32 bits of a generic pointer to LDS == LDS byte offset (ISA 10.2).
    unsigned lds_addr = (unsigned)(uintptr_t)(&stage[threadIdx.x]);

    // CDNA5 async copy: each lane copies 16 bytes from the global address in gsrc to the LDS address in lds_addr.
    asm volatile("global_load_async_to_lds_b128 %0, %1, off"
                 :: "v"(lds_addr), "v"(gsrc)
                 : "memory");

    // Overlap the address math with the in-flight async load.
    const int e   = gid * 4;          // element index (< 48M, fits i32)
    const int row = e / CC;           // 4 | 48 -> a quad never crosses a row
    const int c   = e - row * CC;
    float* dst = fdst + (size_t)parent[row] * CC + c;

    asm volatile("s_wait_asynccnt 0x0" ::: "memory");
    const float4 v = stage[threadIdx.x];   // each lane reads only its own 16B

    unsafeAtomicAdd(dst + 0, v.x);
    unsafeAtomicAdd(dst + 1, v.y);
    unsafeAtomicAdd(dst + 2, v.z);
    unsafeAtomicAdd(dst + 3, v.w);
}

// ---------------------------------------------------------------------------
// Generic upward scatter: fdst[parent[row]] += fsrc[row], quad-granularity.
// Used for f2->f3 and f3->f4 (tables are L2-resident).
// ---------------------------------------------------------------------------
__global__ void __launch_bounds__(256) scatter_up(const float4* __restrict__ fsrc,
                                                  const int* __restrict__ parent,
                                                  float* __restrict__ fdst) {
    const int gid = blockIdx.x * 256 + threadIdx.x;
    const float4 v = fsrc[gid];
    const int e   = gid * 4;
    const int row = e / CC;
    const int c   = e - row * CC;
    float* dst = fdst + (size_t)parent[row] * CC + c;
    unsafeAtomicAdd(dst + 0, v.x);
    unsafeAtomicAdd(dst + 1, v.y);
    unsafeAtomicAdd(dst + 2, v.z);
    unsafeAtomicAdd(dst + 3, v.w);
}

// ---------------------------------------------------------------------------
// Final level as a one-hot GEMM on the f32 WMMA pipe:
//   f5[256 x 48] = S[256 x 2048] * f4[2048 x 48],  S[j,k] = (parent5[k]==j)
// V_WMMA_F32_16X16X4_F32, one 16x16 D-tile per wave.
//   A (16x4 f32, ISA 7.12.2): lanes 0-15 hold M=lane, K={0,1} in v0/v1;
//                             lanes 16-31 hold M=lane-16, K={2,3}.
//   B (4x16 f32, mirrored):   lanes 0-15 hold N=lane, K={0,1};
//                             lanes 16-31 hold N=lane-16, K={2,3}.
//   D (16x16 f32): VGPR r -> M=r (lanes 0-15) / M=r+8 (lanes 16-31), N=lane%16.
// Grid: 3 blocks (N tiles of 16 over C=48) x 512 threads (16 waves = M tiles).
// ---------------------------------------------------------------------------
__global__ void __launch_bounds__(512) f5_wmma(const float* __restrict__ f4,
                                               const int* __restrict__ parent5,
                                               float* __restrict__ f5) {
    __shared__ int p5[N4];
    for (int t = threadIdx.x; t < N4; t += 512) p5[t] = parent5[t];
    __syncthreads();

    const int wave  = threadIdx.x >> 5;      // 0..15 -> M tile
    const int lane  = threadIdx.x & 31;
    const int half  = lane >> 4;             // 0: K={0,1}, 1: K={2,3}
    const int l15   = lane & 15;
    const int nbase = blockIdx.x * 16;       // 0,16,32
    const int mrow  = wave * 16 + l15;       // output row this lane's A covers
    const int ncol  = nbase + l15;

    v8f c = {};
    for (int k = 0; k < N4; k += 4) {
        const int k0 = k + half * 2;
        v2f a, b;
        a.x = (p5[k0]     == mrow) ? 1.0f : 0.0f;
        a.y = (p5[k0 + 1] == mrow) ? 1.0f : 0.0f;
        b.x = f4[(size_t)(k0)     * CC + ncol];
        b.y = f4[(size_t)(k0 + 1) * CC + ncol];
        c = __builtin_amdgcn_wmma_f32_16x16x4_f32(
                /*neg_a=*/false, a, /*neg_b=*/false, b,
                /*c_mod=*/(short)0, c, /*reuse_a=*/false, /*reuse_b=*/false);
    }

#pragma unroll
    for (int r = 0; r < 8; ++r) {
        const int mm = wave * 16 + r + half * 8;
        f5[(size_t)mm * CC + ncol] = c[r];
    }
}

// ---------------------------------------------------------------------------
// Gather + argmax over 48 classes; first-max semantics (strict >) to match
// jnp.argmax. Index written as float (harness output dtype).
// ---------------------------------------------------------------------------
__global__ void __launch_bounds__(256) gather_argmax(const float* __restrict__ f,
                                                     const int* __restrict__ idx,
                                                     float* __restrict__ out,
                                                     int n) {
    const int i = blockIdx.x * 256 + threadIdx.x;
    if (i >= n) return;
    const float4* p = (const float4*)(f + (size_t)idx[i] * CC);  // 192B, 16B-aligned
    float best = -__builtin_huge_valf();
    int bi = 0;
#pragma unroll
    for (int q = 0; q < 12; ++q) {
        const float4 v = p[q];
        if (v.x > best) { best = v.x; bi = q * 4 + 0; }
        if (v.y > best) { best = v.y; bi = q * 4 + 1; }
        if (v.z > best) { best = v.z; bi = q * 4 + 2; }
        if (v.w > best) { best = v.w; bi = q * 4 + 3; }
    }
    out[i] = (float)bi;
}

// ---------------------------------------------------------------------------
extern "C" void kernel_launch(void* const* d_in, const int* in_sizes, int n_in,
                              void* d_out, int out_size, void* d_ws, size_t ws_size,
                              hipStream_t stream) {
    const float* slabel  = (const float*)d_in[0];
    const int*   parent2 = (const int*)d_in[1];
    const int*   parent3 = (const int*)d_in[2];
    const int*   parent4 = (const int*)d_in[3];
    const int*   parent5 = (const int*)d_in[4];
    const int*   idx2    = (const int*)d_in[5];
    const int*   idx3    = (const int*)d_in[6];
    const int*   idx4    = (const int*)d_in[7];
    const int*   idx5    = (const int*)d_in[8];

    float* out = (float*)d_out;
    char*  ws  = (char*)d_ws;
    float* f2 = (float*)(ws);
    float* f3 = (float*)(ws + F2_BYTES);
    float* f4 = (float*)(ws + F2_BYTES + F3_BYTES);
    float* f5 = (float*)(ws + F2_BYTES + F3_BYTES + F4_BYTES);

    // Accumulators must start at zero on every call (graph replay safe).
    hipMemsetAsync(ws, 0, (size_t)WS_TOTAL, stream);

    // Segment-sum hierarchy.
    scatter_l1<<<(N1 * CC / 4) / 256, 256, 0, stream>>>(slabel, parent2, f2);      // 46875 blocks
    scatter_up<<<(N2 * CC / 4) / 256, 256, 0, stream>>>((const float4*)f2, parent3, f3); // 6144
    scatter_up<<<(N3 * CC / 4) / 256, 256, 0, stream>>>((const float4*)f3, parent4, f4); //  768
    f5_wmma  <<<3, 512, 0, stream>>>(f4, parent5, f5);

    // Per-query gathers + argmax, outputs concatenated (s2, s3, s4, s5).
    const int gblocks = (N1 + 255) / 256;
    gather_argmax<<<gblocks, 256, 0, stream>>>(f2, idx2, out + 0 * (size_t)N1, N1);
    gather_argmax<<<gblocks, 256, 0, stream>>>(f3, idx3, out + 1 * (size_t)N1, N1);
    gather_argmax<<<gblocks, 256, 0, stream>>>(f4, idx4, out + 2 * (size_t)N1, N1);
    gather_argmax<<<gblocks, 256, 0, stream>>>(f5, idx5, out + 3 * (size_t)N1, N1);
}